// LSAM_55250459295991
// MI455X (gfx1250) — compile-verified
//
#include <hip/hip_runtime.h>
#include <hip/hip_bf16.h>

typedef __attribute__((ext_vector_type(2))) float v2f;
typedef __attribute__((ext_vector_type(8))) float v8f;

// D(16x16 f32) = A(16x4 f32) * B(4x16 f32) + C ; full-precision CDNA5 WMMA
static __device__ __forceinline__ v8f wmma4(v2f a, v2f b, v8f c) {
  return __builtin_amdgcn_wmma_f32_16x16x4_f32(false, a, false, b, (short)0, c, false, false);
}

#define NB   8
#define C_   256
#define HW_  160
#define HP_  161
#define NWH  23          // windows per side
#define P_   49          // positions per window
#define FROW 1127        // 23*49, flat offset per window-row in the "view"
#define XST  260         // LDS stride for X window slab [j][ci]
#define QST  68          // LDS stride for 64-col tiles
#define SPART_FLOATS (NB * NWH * P_ * P_)   // 441784

// -------- Kernel 1: partial attention logits  S[p,q] = sum_{ch,win} Q K ------
__global__ __launch_bounds__(256) void k_qk_logits(
    const float* __restrict__ x,  const float* __restrict__ Wq,
    const float* __restrict__ bq, const float* __restrict__ Wk,
    const float* __restrict__ bk, float* __restrict__ Spart)
{
  __shared__ float Xt[P_ * XST];    // X window, transposed: [j][ci]
  __shared__ float QKT[64 * QST];   // QK^T: [p][ch64]  (cols 0..31 Q, 32..63 K)

  const int ih   = blockIdx.x;          // window row 0..22
  const int b    = blockIdx.y;
  const int tid  = threadIdx.x;
  const int lane = tid & 31;
  const int wave = tid >> 5;
  const int LL   = lane & 15;
  const int hi   = lane >> 4;           // half-wave select (K pairs {0,1}/{2,3})
  const int koff = hi * 2;

  // GEMM1 tiles: 64x64 output, 16 tiles over 8 waves (2 n-tiles each)
  const int mt  = wave & 3;
  const int nt0 = (wave >> 2) * 2, nt1 = nt0 + 1;
  const int arow = mt * 16 + LL;        // stacked [Wq;Wk] row
  const float* wrowp = (arow < 32) ? (Wq + arow * 256) : (Wk + (arow - 32) * 256);
  const int n0 = min(nt0 * 16 + LL, 48);   // clamp: cols >=49 become garbage,
  const int n1 = min(nt1 * 16 + LL, 48);   // discarded downstream

  // GEMM2 tiles: 49x49 (padded 64x64), persistent accumulation over 23 windows
  const int pt  = wave & 3;
  const int qt0 = (wave >> 2) * 2, qt1 = qt0 + 1;
  v8f s0 = {}, s1 = {};

  for (int iw = 0; iw < NWH; ++iw) {
    __syncthreads();
    // gather window (reflect pad: 160 -> 158) into LDS, transposed
    for (int idx = tid; idx < C_ * P_; idx += 256) {
      int j = idx % P_, ci = idx / P_;
      int yy = ih * 7 + j / 7, xx = iw * 7 + j % 7;
      int y  = (yy < HW_) ? yy : 158;
      int xc = (xx < HW_) ? xx : 158;
      Xt[j * XST + ci] = x[(((size_t)b * C_ + ci) * HW_ + y) * HW_ + xc];
    }
    __syncthreads();

    // GEMM1: QK = [Wq;Wk](64x256) @ Xwin(256x49)
    v8f a0 = {}, a1 = {};
    #pragma unroll 4
    for (int k4 = 0; k4 < 64; ++k4) {
      v2f af  = *(const v2f*)(wrowp + k4 * 4 + koff);
      v2f bf0 = *(const v2f*)&Xt[n0 * XST + k4 * 4 + koff];
      v2f bf1 = *(const v2f*)&Xt[n1 * XST + k4 * 4 + koff];
      a0 = wmma4(af, bf0, a0);
      a1 = wmma4(af, bf1, a1);
    }
    // QK^T (+bias) into LDS
    #pragma unroll
    for (int r = 0; r < 8; ++r) {
      int ch = mt * 16 + r + hi * 8;
      float bias = (ch < 32) ? bq[ch] : bk[ch - 32];
      QKT[(nt0 * 16 + LL) * QST + ch] = a0[r] + bias;
      QKT[(nt1 * 16 + LL) * QST + ch] = a1[r] + bias;
    }
    __syncthreads();

    // GEMM2: S += Q^T(49x32) @ K(32x49)
    #pragma unroll
    for (int k4 = 0; k4 < 8; ++k4) {
      v2f af  = *(const v2f*)&QKT[(pt  * 16 + LL) * QST + k4 * 4 + koff];
      v2f bf0 = *(const v2f*)&QKT[(qt0 * 16 + LL) * QST + 32 + k4 * 4 + koff];
      v2f bf1 = *(const v2f*)&QKT[(qt1 * 16 + LL) * QST + 32 + k4 * 4 + koff];
      s0 = wmma4(af, bf0, s0);
      s1 = wmma4(af, bf1, s1);
    }
  }

  // deterministic two-phase reduction: write this group's partial
  float* sp = Spart + ((size_t)b * NWH + ih) * (P_ * P_);
  #pragma unroll
  for (int r = 0; r < 8; ++r) {
    int p  = pt * 16 + r + hi * 8;
    int q0 = qt0 * 16 + LL, q1 = qt1 * 16 + LL;
    if (p < P_) {
      if (q0 < P_) sp[p * P_ + q0] = s0[r];
      if (q1 < P_) sp[p * P_ + q1] = s1[r];
    }
  }
}

// -------- Kernel 2: reduce partials + row softmax -> attn[b,49,49] ----------
__global__ __launch_bounds__(256) void k_softmax(const float* __restrict__ Spart,
                                                 float* __restrict__ attn)
{
  const int b = blockIdx.x;
  const int lane = threadIdx.x & 31;
  const int wave = threadIdx.x >> 5;
  for (int p = wave; p < P_; p += 8) {
    int q0 = lane, q1 = lane + 32;
    bool v1 = (q1 < P_);
    float s0 = 0.f, s1 = 0.f;
    for (int g = 0; g < NWH; ++g) {
      const float* sp = Spart + ((size_t)b * NWH + g) * (P_ * P_) + p * P_;
      s0 += sp[q0];
      if (v1) s1 += sp[q1];
    }
    float m = v1 ? fmaxf(s0, s1) : s0;
    #pragma unroll
    for (int off = 16; off > 0; off >>= 1) m = fmaxf(m, __shfl_xor(m, off, 32));
    float e0 = expf(s0 - m);
    float e1 = v1 ? expf(s1 - m) : 0.f;
    float t = e0 + e1;
    #pragma unroll
    for (int off = 16; off > 0; off >>= 1) t += __shfl_xor(t, off, 32);
    float inv = 1.0f / t;
    attn[(size_t)b * (P_ * P_) + p * P_ + q0] = e0 * inv;
    if (v1) attn[(size_t)b * (P_ * P_) + p * P_ + q1] = e1 * inv;
  }
}

// -------- Kernel 3: V conv + out = V @ attn^T + view/crop scatter -----------
__global__ __launch_bounds__(256) void k_vout(
    const float* __restrict__ x, const float* __restrict__ Wv,
    const float* __restrict__ bv, const float* __restrict__ attn,
    float* __restrict__ out)
{
  __shared__ float Xt[P_ * XST];     // [j][ci]
  __shared__ float VL[128 * QST];    // half the channels of V: [c_local][j]
  __shared__ float AT[64 * QST];     // attn, zero-padded to 64x64: [i][j]

  const int win = blockIdx.x;        // 0..528
  const int b   = blockIdx.y;
  const int ih  = win / NWH, iw = win % NWH;
  const int tid = threadIdx.x;
  const int lane = tid & 31, wave = tid >> 5;
  const int LL = lane & 15, hi = lane >> 4, koff = hi * 2;

  for (int idx = tid; idx < 64 * QST; idx += 256) AT[idx] = 0.f;
  __syncthreads();
  for (int idx = tid; idx < P_ * P_; idx += 256)
    AT[(idx / P_) * QST + (idx % P_)] = attn[(size_t)b * (P_ * P_) + idx];
  for (int idx = tid; idx < C_ * P_; idx += 256) {
    int j = idx % P_, ci = idx / P_;
    int yy = ih * 7 + j / 7, xx = iw * 7 + j % 7;
    int y  = (yy < HW_) ? yy : 158;
    int xc = (xx < HW_) ? xx : 158;
    Xt[j * XST + ci] = x[(((size_t)b * C_ + ci) * HW_ + y) * HW_ + xc];
  }
  __syncthreads();

  int nc0 = min(0 * 16 + LL, 48), nc1 = min(1 * 16 + LL, 48);
  int nc2 = min(2 * 16 + LL, 48), nc3 = min(3 * 16 + LL, 48);
  const int fbase = ih * FROW + iw * P_;

  for (int half = 0; half < 2; ++half) {
    // GEMM-V: 128 channels of  V = Wv @ Xwin
    const int ct = half * 8 + wave;                 // global m-tile
    const float* wrow = Wv + (ct * 16 + LL) * 256;
    v8f a0 = {}, a1 = {}, a2 = {}, a3 = {};
    #pragma unroll 4
    for (int k4 = 0; k4 < 64; ++k4) {
      v2f af = *(const v2f*)(wrow + k4 * 4 + koff);
      v2f b0 = *(const v2f*)&Xt[nc0 * XST + k4 * 4 + koff];
      v2f b1 = *(const v2f*)&Xt[nc1 * XST + k4 * 4 + koff];
      v2f b2 = *(const v2f*)&Xt[nc2 * XST + k4 * 4 + koff];
      v2f b3 = *(const v2f*)&Xt[nc3 * XST + k4 * 4 + koff];
      a0 = wmma4(af, b0, a0);  a1 = wmma4(af, b1, a1);
      a2 = wmma4(af, b2, a2);  a3 = wmma4(af, b3, a3);
    }
    #pragma unroll
    for (int r = 0; r < 8; ++r) {
      int cl = wave * 16 + r + hi * 8;
      float bias = bv[half * 128 + cl];
      VL[cl * QST +  0 + LL] = a0[r] + bias;
      VL[cl * QST + 16 + LL] = a1[r] + bias;
      VL[cl * QST + 32 + LL] = a2[r] + bias;
      VL[cl * QST + 48 + LL] = a3[r] + bias;
    }
    __syncthreads();

    // GEMM-O: Out[c,i] = sum_j V[c,j] * attn[i,j]  (K = 64 padded; AT zero for j>=49)
    v8f o0 = {}, o1 = {}, o2 = {}, o3 = {};
    #pragma unroll
    for (int k4 = 0; k4 < 16; ++k4) {
      v2f af = *(const v2f*)&VL[(wave * 16 + LL) * QST + k4 * 4 + koff];
      v2f b0 = *(const v2f*)&AT[( 0 + LL) * QST + k4 * 4 + koff];
      v2f b1 = *(const v2f*)&AT[(16 + LL) * QST + k4 * 4 + koff];
      v2f b2 = *(const v2f*)&AT[(32 + LL) * QST + k4 * 4 + koff];
      v2f b3 = *(const v2f*)&AT[(48 + LL) * QST + k4 * 4 + koff];
      o0 = wmma4(af, b0, o0);  o1 = wmma4(af, b1, o1);
      o2 = wmma4(af, b2, o2);  o3 = wmma4(af, b3, o3);
    }
    // scatter through the contiguous-view remap + crop
    #pragma unroll
    for (int r = 0; r < 8; ++r) {
      int ch = half * 128 + wave * 16 + r + hi * 8;
      float* outc = out + ((size_t)b * C_ + ch) * (HW_ * HW_);
      float vals[4] = {o0[r], o1[r], o2[r], o3[r]};
      #pragma unroll
      for (int it = 0; it < 4; ++it) {
        int i = it * 16 + LL;
        if (i < P_) {
          int f = fbase + i;
          int Y = f / HP_;
          int X = f - Y * HP_;
          if (Y < HW_ && X < HW_) outc[Y * HW_ + X] = vals[it];
        }
      }
    }
    __syncthreads();
  }
}

extern "C" void kernel_launch(void* const* d_in, const int* in_sizes, int n_in,
                              void* d_out, int out_size, void* d_ws, size_t ws_size,
                              hipStream_t stream) {
  (void)in_sizes; (void)n_in; (void)out_size; (void)ws_size;
  const float* x  = (const float*)d_in[0];
  const float* Wq = (const float*)d_in[1];
  const float* bq = (const float*)d_in[2];
  const float* Wk = (const float*)d_in[3];
  const float* bk = (const float*)d_in[4];
  const float* Wv = (const float*)d_in[5];
  const float* bv = (const float*)d_in[6];
  float* out   = (float*)d_out;
  float* Spart = (float*)d_ws;                 // [8][23][49*49]
  float* attn  = Spart + SPART_FLOATS;         // [8][49*49]

  k_qk_logits<<<dim3(NWH, NB), 256, 0, stream>>>(x, Wq, bq, Wk, bk, Spart);
  k_softmax  <<<dim3(NB),      256, 0, stream>>>(Spart, attn);
  k_vout     <<<dim3(NWH * NWH, NB), 256, 0, stream>>>(x, Wv, bv, attn, out);
}